// HeteroGNN_12051678233154
// MI455X (gfx1250) — compile-verified
//
#include <hip/hip_runtime.h>

#define N_NODES    100000
#define N_EDGES_K  1000000
#define K_ETYPES   4
#define IN_F       128
#define OUT_F      64
// K-pair-interleaved LDS layout: pair-row p holds (W[2p][n], W[2p+1][n]) adjacent.
// Stride 160 floats => 160 mod 64 = 32, so pair-rows p and p+1 occupy
// complementary halves of the 64 banks -> conflict-free ds_load_b64 per half-wave.
#define PSTRIDE    160
#define N_PAIRS    (IN_F / 2)   // 64 pair rows

typedef __attribute__((ext_vector_type(2))) float v2f;
typedef __attribute__((ext_vector_type(8))) float v8f;

// ---------------- small helper kernels ----------------

__global__ __launch_bounds__(256) void zero_f32(float* __restrict__ p, int n) {
    int i = blockIdx.x * 256 + threadIdx.x;
    if (i < n) p[i] = 0.0f;
}

__global__ __launch_bounds__(256) void degree_count(const int* __restrict__ src,
                                                    const int* __restrict__ dst,
                                                    float* __restrict__ deg_src,
                                                    float* __restrict__ deg_dst,
                                                    int n_edges) {
    int e = blockIdx.x * 256 + threadIdx.x;
    if (e < n_edges) {
        atomicAdd(&deg_src[src[e]], 1.0f);
        atomicAdd(&deg_dst[dst[e]], 1.0f);
    }
}

__global__ __launch_bounds__(256) void deg_to_norm(float* __restrict__ p, int n) {
    int i = blockIdx.x * 256 + threadIdx.x;
    if (i < n) p[i] = rsqrtf(fmaxf(p[i], 1.0f));
}

// out[n,f] = sum_k b[k,f]   (bias contributions, done once)
__global__ __launch_bounds__(256) void init_bias(const float* __restrict__ b,
                                                 float* __restrict__ out) {
    int i = blockIdx.x * 256 + threadIdx.x;
    if (i < N_NODES * OUT_F) {
        int f = i & (OUT_F - 1);
        out[i] = b[f] + b[OUT_F + f] + b[2 * OUT_F + f] + b[3 * OUT_F + f];
    }
}

// ---------------- WMMA GEMM: feat = (h * norm_src) @ Wk ----------------
// 8 waves per WG; each wave: one 16(M) x 64(N) tile, K=128 in steps of 4
// via V_WMMA_F32_16X16X4_F32 (exact fp32 matrix path on CDNA5).
// B operands come from K-pair-interleaved LDS as single 8-byte loads.
__global__ __launch_bounds__(256) void gemm_feat(const float* __restrict__ h,
                                                 const float* __restrict__ Wk,
                                                 const float* __restrict__ norm_src,
                                                 float* __restrict__ feat) {
    __shared__ float Wl[N_PAIRS * PSTRIDE];   // 40 KB

    int tid = threadIdx.x;
    // cooperative load of W[k] (128x64 f32) into pair-interleaved LDS:
    // Wl[p*PSTRIDE + n*2 + 0] = W[2p][n],  +1 = W[2p+1][n]
    for (int i = tid; i < N_PAIRS * OUT_F; i += 256) {
        int p = i >> 6;          // pair row
        int n = i & (OUT_F - 1); // column
        float2 w;
        w.x = Wk[(2 * p) * OUT_F + n];      // coalesced across n
        w.y = Wk[(2 * p + 1) * OUT_F + n];
        *(float2*)(&Wl[p * PSTRIDE + n * 2]) = w;
    }
    __syncthreads();

    int wave    = tid >> 5;
    int lane    = tid & 31;
    int halfsel = lane >> 4;      // 0 = lanes 0-15, 1 = lanes 16-31
    int lo      = lane & 15;

    int mtile = blockIdx.x * 8 + wave;
    int row   = mtile * 16 + lo;
    int rc    = row < N_NODES ? row : (N_NODES - 1);   // clamp: EXEC must stay all-1s
    float ns  = norm_src[rc];
    const float* hrow = h + rc * IN_F;
    int khalf = halfsel * 2;      // this half's K offset within the 4-wide step

    v8f acc0 = {}, acc1 = {}, acc2 = {}, acc3 = {};

    for (int k0 = 0; k0 < IN_F; k0 += 4) {
        // A (16x4 f32): lanes 0-15 carry K=k0,k0+1 of row M=lo; lanes 16-31 K=k0+2,k0+3
        float2 hv = *(const float2*)(hrow + k0 + khalf);
        v2f a; a.x = hv.x * ns; a.y = hv.y * ns;

        // B (4x16 f32): pair-row p supplies (K=2p, K=2p+1) for column lo+16j in one b64 load
        int p = (k0 + khalf) >> 1;          // k0/2 for lanes 0-15, +1 for lanes 16-31
        const float* wp = &Wl[p * PSTRIDE + lo * 2];
        v2f b0 = *(const v2f*)(wp + 0);
        v2f b1 = *(const v2f*)(wp + 32);
        v2f b2 = *(const v2f*)(wp + 64);
        v2f b3 = *(const v2f*)(wp + 96);

        acc0 = __builtin_amdgcn_wmma_f32_16x16x4_f32(false, a, false, b0, (short)0, acc0, false, false);
        acc1 = __builtin_amdgcn_wmma_f32_16x16x4_f32(false, a, false, b1, (short)0, acc1, false, false);
        acc2 = __builtin_amdgcn_wmma_f32_16x16x4_f32(false, a, false, b2, (short)0, acc2, false, false);
        acc3 = __builtin_amdgcn_wmma_f32_16x16x4_f32(false, a, false, b3, (short)0, acc3, false, false);
    }

    // C/D layout: VGPR v -> M = v (lanes 0-15) / v+8 (lanes 16-31); N = lo + 16j
    int mbase = mtile * 16 + halfsel * 8;
#pragma unroll
    for (int v = 0; v < 8; ++v) {
        int m = mbase + v;
        if (m < N_NODES) {
            float* o = feat + m * OUT_F + lo;
            o[0]  = acc0[v];
            o[16] = acc1[v];
            o[32] = acc2[v];
            o[48] = acc3[v];
        }
    }
}

// ---------------- edge scatter: out[dst] += feat[src] * norm_dst[dst] ----------------
// 16 threads per edge, float4 gather + 4 f32 atomics (L2-resident: feat & out ~25.6 MB each)
__global__ __launch_bounds__(256) void scatter_add(const float* __restrict__ feat,
                                                   const int* __restrict__ src,
                                                   const int* __restrict__ dst,
                                                   const float* __restrict__ norm_dst,
                                                   float* __restrict__ out,
                                                   int n_edges) {
    int gid = blockIdx.x * 256 + threadIdx.x;
    int e = gid >> 4;
    if (e >= n_edges) return;
    int g = (gid & 15) * 4;
    int s = src[e], d = dst[e];
    float w = norm_dst[d];
    float4 f = *(const float4*)(feat + s * OUT_F + g);
    float* o = out + d * OUT_F + g;
    atomicAdd(o + 0, f.x * w);
    atomicAdd(o + 1, f.y * w);
    atomicAdd(o + 2, f.z * w);
    atomicAdd(o + 3, f.w * w);
}

// ---------------- launch ----------------

extern "C" void kernel_launch(void* const* d_in, const int* in_sizes, int n_in,
                              void* d_out, int out_size, void* d_ws, size_t ws_size,
                              hipStream_t stream) {
    const float* h   = (const float*)d_in[0];  // [N_NODES, 128]
    const float* W   = (const float*)d_in[1];  // [4, 128, 64]
    const float* b   = (const float*)d_in[2];  // [4, 64]
    const int*   src = (const int*)d_in[3];    // [4, 1M]
    const int*   dst = (const int*)d_in[4];    // [4, 1M]
    float* out = (float*)d_out;                // [N_NODES, 64]

    char* ws = (char*)d_ws;
    float* feat    = (float*)ws;                                    // 25.6 MB
    float* deg     = (float*)(ws + (size_t)N_NODES * OUT_F * 4);    // 0.8 MB (src|dst)
    float* deg_src = deg;
    float* deg_dst = deg + N_NODES;

    // out = sum_k b[k]  (broadcast per node)
    init_bias<<<(N_NODES * OUT_F + 255) / 256, 256, 0, stream>>>(b, out);

    const int m_tiles = (N_NODES + 15) / 16;          // 6250
    const int gemm_wgs = (m_tiles + 7) / 8;           // 782

    for (int k = 0; k < K_ETYPES; ++k) {
        const int*   sk = src + (size_t)k * N_EDGES_K;
        const int*   dk = dst + (size_t)k * N_EDGES_K;
        const float* Wk = W + (size_t)k * IN_F * OUT_F;

        zero_f32<<<(2 * N_NODES + 255) / 256, 256, 0, stream>>>(deg, 2 * N_NODES);
        degree_count<<<(N_EDGES_K + 255) / 256, 256, 0, stream>>>(sk, dk, deg_src, deg_dst, N_EDGES_K);
        deg_to_norm<<<(2 * N_NODES + 255) / 256, 256, 0, stream>>>(deg, 2 * N_NODES);

        gemm_feat<<<gemm_wgs, 256, 0, stream>>>(h, Wk, deg_src, feat);

        scatter_add<<<(N_EDGES_K * 16 + 255) / 256, 256, 0, stream>>>(feat, sk, dk, deg_dst, out, N_EDGES_K);
    }
}